// DDK_77644418777653
// MI455X (gfx1250) — compile-verified
//
#include <hip/hip_runtime.h>
#include <cstdint>

// Problem geometry (fixed by the reference).
#define M_ROWS 262144
#define N_COLS 512
#define CHUNK 64
#define NCHUNK (N_COLS / CHUNK) // 8
#define ROWS_PER_WAVE 32        // one row per lane (wave32)
#define WAVES_PER_BLOCK 2
#define BLOCK_THREADS (WAVES_PER_BLOCK * 32)
#define ROWSTRIDE 65 // floats; bank = (65*lane + k) % 64 = (lane+k)%64 -> conflict-free

#if __has_builtin(__builtin_amdgcn_tensor_load_to_lds) && \
    __has_builtin(__builtin_amdgcn_tensor_store_from_lds)
#define USE_TDM 1
#else
#define USE_TDM 0
#endif

typedef unsigned int v4u __attribute__((ext_vector_type(4)));
typedef int v4i __attribute__((ext_vector_type(4)));
typedef int v8i __attribute__((ext_vector_type(8)));

#if USE_TDM
// LDS byte offset of a __shared__ pointer (addrspace(3) pointers hold the raw
// LDS offset; hardware adds the wave's LDS_BASE).
__device__ __forceinline__ unsigned lds_addr_of(void* p) {
  return (unsigned)(unsigned long long)(__attribute__((address_space(3))) char*)p;
}

// D# group 0: count=1 (valid), lds_addr, 57-bit global_addr, type=2 ("image").
__device__ __forceinline__ v4u make_g0(unsigned lds, unsigned long long ga) {
  v4u g0;
  g0.x = 1u;                                                    // count=1, user mode
  g0.y = lds;                                                   // lds_addr (bytes)
  g0.z = (unsigned)ga;                                          // global_addr[31:0]
  g0.w = (unsigned)((ga >> 32) & 0x01FFFFFFull) | 0x80000000u;  // addr[56:32] | type=2
  return g0;
}

__device__ __forceinline__ void tdm_load(unsigned lds, unsigned long long ga,
                                         v8i g1, v4i g2, v4i g3) {
#if defined(__clang_major__) && (__clang_major__ >= 23)
  v8i z8 = {0, 0, 0, 0, 0, 0, 0, 0};
  __builtin_amdgcn_tensor_load_to_lds(make_g0(lds, ga), g1, g2, g3, z8, 0);
#else
  __builtin_amdgcn_tensor_load_to_lds(make_g0(lds, ga), g1, g2, g3, 0);
#endif
}

__device__ __forceinline__ void tdm_store(unsigned lds, unsigned long long ga,
                                          v8i g1, v4i g2, v4i g3) {
#if defined(__clang_major__) && (__clang_major__ >= 23)
  v8i z8 = {0, 0, 0, 0, 0, 0, 0, 0};
  __builtin_amdgcn_tensor_store_from_lds(make_g0(lds, ga), g1, g2, g3, z8, 0);
#else
  __builtin_amdgcn_tensor_store_from_lds(make_g0(lds, ga), g1, g2, g3, 0);
#endif
}
#endif // USE_TDM

__global__ __launch_bounds__(BLOCK_THREADS) void DDK_recur_scan_kernel(
    const float* __restrict__ in, const float* __restrict__ alpha,
    const float* __restrict__ beta, float* __restrict__ out) {
  // Two double-buffered 32x(64+pad) f32 tiles, one pair per wave. Waves are
  // fully independent: no __syncthreads anywhere.
  __shared__ float tile[WAVES_PER_BLOCK][2][ROWS_PER_WAVE * ROWSTRIDE];

  const int lane = (int)(threadIdx.x & 31u);
  const int wave = __builtin_amdgcn_readfirstlane((int)(threadIdx.x >> 5));
  const long long rowBase =
      (long long)blockIdx.x * (WAVES_PER_BLOCK * ROWS_PER_WAVE) +
      (long long)wave * ROWS_PER_WAVE;
  const float* gin = in + rowBase * N_COLS;
  float* gout = out + rowBase * N_COLS;

  const float a = alpha[0];
  const float b = beta[0];
  float w = 1.0f; // w[:,0] == 1

  float* buf0 = &tile[wave][0][0];
  float* buf1 = &tile[wave][1][0];

#if USE_TDM
  // ---- D# group 1 for loads: 2D tile 32x64 f32, LDS pad 1 dword per 64 ----
  v8i g1ld;
  g1ld[0] = (2 << 16)    // data_size = 4 bytes
          | (1 << 20)    // pad_enable (load only)
          | (5 << 22)    // pad_interval: 64 dwords
          | (0 << 25);   // pad_amount: 1 dword  -> LDS row stride = 65 floats
  g1ld[1] = (CHUNK & 0xFFFF) << 16;          // tensor_dim0 = 64 (extent from tile start)
  g1ld[2] = (ROWS_PER_WAVE & 0xFFFF) << 16;  // tensor_dim1 = 32
  g1ld[3] = (CHUNK & 0xFFFF) << 16;          // tile_dim0 = 64
  g1ld[4] = ROWS_PER_WAVE;                   // tile_dim1 = 32, tile_dim2 = 0
  g1ld[5] = N_COLS;                          // tensor_dim0_stride = 512
  g1ld[6] = 0;
  g1ld[7] = 0;

  // ---- D# group 1/2 for stores: iterate mode, 32 x (1x64) row tiles -------
  v8i g1st;
  g1st[0] = (2 << 16) | (1 << 19);           // data_size = 4B, iterate_enable
  g1st[1] = (CHUNK & 0xFFFF) << 16;          // tensor_dim0 = 64
  g1st[2] = (1) << 16;                       // tensor_dim1 = 1
  g1st[3] = (CHUNK & 0xFFFF) << 16;          // tile_dim0 = 64
  g1st[4] = 1;                               // tile_dim1 = 1
  g1st[5] = N_COLS;
  g1st[6] = 0;
  g1st[7] = 0;
  v4i g2st;
  g2st[0] = 0;                               // tensor_dim2 unused
  g2st[1] = ROWSTRIDE;                       // lds_addr_increment   = 65 elems
  g2st[2] = N_COLS;                          // global_addr_increment = 512 elems
  g2st[3] = (ROWS_PER_WAVE - 1) << 16;       // iterate_count = 31 -> 32 iterations
  v4i gz = {0, 0, 0, 0};

  const unsigned lds0 = lds_addr_of(buf0);
  const unsigned lds1 = lds_addr_of(buf1);

  // Prime the pipeline with chunk 0.
  tdm_load(lds0, (unsigned long long)(uintptr_t)gin, g1ld, gz, gz);

  for (int c = 0; c < NCHUNK; ++c) {
    float* buf = (c & 1) ? buf1 : buf0;
    const unsigned ldsb = (c & 1) ? lds1 : lds0;

    // Prefetch next chunk into the other buffer. Safe: this wave's TDM ops
    // complete in order, so store(c-1) (which reads that buffer) finishes
    // before load(c+1) overwrites it.
    if (c + 1 < NCHUNK) {
      tdm_load((c & 1) ? lds0 : lds1,
               (unsigned long long)(uintptr_t)(gin + (c + 1) * CHUNK),
               g1ld, gz, gz);
      __builtin_amdgcn_s_wait_tensorcnt(1); // load(c) (and store(c-1)) done
    } else {
      __builtin_amdgcn_s_wait_tensorcnt(0);
    }
    __asm__ volatile("" ::: "memory");

    // Serial recurrence for this lane's row; in-place: out value replaces x.
    float* row = buf + lane * ROWSTRIDE;
#pragma unroll
    for (int k = 0; k < CHUNK; ++k) {
      float x = row[k];
      row[k] = w;             // out[row, c*64+k] = w_{c*64+k}
      w = w + a / w - b * x;  // advance (last chunk's final advance is discarded)
    }

    // Drain this wave's LDS writes before the TDM store reads them.
#if __has_builtin(__builtin_amdgcn_s_wait_dscnt)
    __asm__ volatile("" ::: "memory");
    __builtin_amdgcn_s_wait_dscnt(0);
#else
    __asm__ volatile("s_wait_dscnt 0" ::: "memory");
#endif
    tdm_store(ldsb, (unsigned long long)(uintptr_t)(gout + c * CHUNK),
              g1st, g2st, gz);
  }
  // s_endpgm performs an implicit wait-idle, covering the final TDM store.

#else // ------------------- fallback: vector-pipe staging -------------------
  for (int c = 0; c < NCHUNK; ++c) {
    float* buf = (c & 1) ? buf1 : buf0;
    // Coalesced load: 16 lanes cover one row-segment of 16 float4 (256B).
#pragma unroll
    for (int i = 0; i < 16; ++i) {
      int f = lane + 32 * i;
      int r = f >> 4, c4 = f & 15;
      const float4 v =
          *(const float4*)(gin + (long long)r * N_COLS + c * CHUNK + c4 * 4);
      float* d = buf + r * ROWSTRIDE + c4 * 4;
      d[0] = v.x; d[1] = v.y; d[2] = v.z; d[3] = v.w;
    }
    // Intra-wave only: LDS is in-order per wave, no barrier needed.
    float* row = buf + lane * ROWSTRIDE;
#pragma unroll
    for (int k = 0; k < CHUNK; ++k) {
      float x = row[k];
      row[k] = w;
      w = w + a / w - b * x;
    }
#pragma unroll
    for (int i = 0; i < 16; ++i) {
      int f = lane + 32 * i;
      int r = f >> 4, c4 = f & 15;
      const float* s = buf + r * ROWSTRIDE + c4 * 4;
      float4 v;
      v.x = s[0]; v.y = s[1]; v.z = s[2]; v.w = s[3];
      *(float4*)(gout + (long long)r * N_COLS + c * CHUNK + c4 * 4) = v;
    }
  }
#endif
}

extern "C" void kernel_launch(void* const* d_in, const int* in_sizes, int n_in,
                              void* d_out, int out_size, void* d_ws,
                              size_t ws_size, hipStream_t stream) {
  (void)in_sizes; (void)n_in; (void)d_ws; (void)ws_size; (void)out_size;
  const float* in = (const float*)d_in[0];
  const float* alpha = (const float*)d_in[1];
  const float* beta = (const float*)d_in[2];
  float* out = (float*)d_out;

  const int rows_per_block = WAVES_PER_BLOCK * ROWS_PER_WAVE; // 64
  dim3 grid(M_ROWS / rows_per_block);                         // 4096
  dim3 block(BLOCK_THREADS);                                  // 64 (2 waves)
  DDK_recur_scan_kernel<<<grid, block, 0, stream>>>(in, alpha, beta, out);
}